// GRCUvH_2637109920162
// MI455X (gfx1250) — compile-verified
//
#include <hip/hip_runtime.h>
#include <math.h>

typedef __attribute__((ext_vector_type(2))) float v2f;
typedef __attribute__((ext_vector_type(8))) float v8f;
typedef int v4i __attribute__((vector_size(16)));
#define ASG __attribute__((address_space(1)))
#define ASL __attribute__((address_space(3)))

#define T_STEPS 4
#define N_NODES 20000
#define N_EDGES 640000
#define D_IN    256
#define D_OUT   128
#define KSEL    128
#define SLOPE   0.22916666666666666f  // (1/8 + 1/3)/2

// ---- workspace layout (float offsets) ----
#define OFF_PNORM  0
#define OFF_CNT    16        // 16 u32 counters
#define OFF_HIST   32        // 2048 u32
#define OFF_SCORES 2080      // 20000 f
#define OFF_TVALS  22080     // 128 f
#define OFF_TIDX   22208     // 128 i32
#define OFF_XT     22336     // 256*128 f (paired layout)
#define OFF_Z      55104     // 256*128 f (paired layout)
#define OFF_R      87872     // 256*128 f (paired layout)
#define OFF_W0B    120640    // 256*128 f (paired layout)
#define OFF_W1B    153408    // 256*128 f (paired layout)
#define OFF_Y      186176    // 20000*128 f
#define OFF_AGG    2746176   // 20000*128 f  (total ~21.2 MB)

// paired layout for K-indexed [256 x 128] operands: element (k,c) lives at
// P(k,c) = (k>>1)*256 + 2c + (k&1)  -> pair (k,k+1) at column c is contiguous (8B)
__device__ __forceinline__ size_t pidx(int k, int c) {
    return (size_t)(k >> 1) * 256 + ((size_t)c << 1) + (k & 1);
}

__device__ __forceinline__ unsigned fkey(float f) {
    unsigned u = __float_as_uint(f);
    return (u & 0x80000000u) ? ~u : (u | 0x80000000u);
}

__device__ __forceinline__ v8f wmma4(v2f a, v2f b, v8f c) {
    // V_WMMA_F32_16X16X4_F32: D = A(16x4 f32) x B(4x16 f32) + C(16x16 f32)
    return __builtin_amdgcn_wmma_f32_16x16x4_f32(false, a, false, b, (short)0, c, false, false);
}

// ---------------- utility ----------------
__global__ void k_zero_f4(float4* __restrict__ p, int n4) {
    int i = blockIdx.x * blockDim.x + threadIdx.x;
    if (i < n4) p[i] = make_float4(0.f, 0.f, 0.f, 0.f);
}

// row-major [256][128] -> paired layout
__global__ void k_pack_pairs(const float* __restrict__ src, float* __restrict__ dst) {
    int i = blockIdx.x * blockDim.x + threadIdx.x;   // 0 .. 32767
    int row = i >> 7, col = i & 127;
    dst[pidx(row, col)] = src[i];
}

__global__ void k_pnorm(const float* __restrict__ p, float* __restrict__ pn) {
    __shared__ float s[256];
    float v = p[threadIdx.x];
    s[threadIdx.x] = v * v;
    __syncthreads();
    for (int o = 128; o > 0; o >>= 1) {
        if (threadIdx.x < o) s[threadIdx.x] += s[threadIdx.x + o];
        __syncthreads();
    }
    if (threadIdx.x == 0) pn[0] = sqrtf(s[0]);
}

// ---------------- scores: (x @ p)/||p|| + mask ----------------
__global__ void k_scores(const float* __restrict__ x, const float* __restrict__ p,
                         const float* __restrict__ mask, const float* __restrict__ pn,
                         float* __restrict__ scores) {
    int wave = threadIdx.x >> 5, lane = threadIdx.x & 31;
    int row = blockIdx.x * 8 + wave;
    if (row >= N_NODES) return;
    const float* xr = x + (size_t)row * D_IN;
    float acc = 0.f;
    for (int d = lane; d < D_IN; d += 32) acc += xr[d] * p[d];
    for (int o = 16; o > 0; o >>= 1) acc += __shfl_down(acc, o, 32);
    if (lane == 0) scores[row] = acc / pn[0] + mask[row];
}

// ---------------- radix-select top-128 ----------------
__global__ void k_hist(const float* __restrict__ scores, unsigned* __restrict__ hist) {
    int n = blockIdx.x * blockDim.x + threadIdx.x;
    if (n < N_NODES) atomicAdd(&hist[fkey(scores[n]) >> 21], 1u);
}

__global__ void k_select(const unsigned* __restrict__ hist, unsigned* __restrict__ cnt) {
    if (threadIdx.x == 0) {
        unsigned acc = 0; int thr = 0;
        for (int b = 2047; b >= 0; --b) {
            unsigned c = hist[b];
            if (acc + c >= (unsigned)KSEL) { thr = b; break; }
            acc += c;
        }
        cnt[0] = (unsigned)thr;   // threshold bucket
        cnt[1] = acc;             // count strictly above bucket
        cnt[4] = KSEL - acc;      // remaining to take from threshold bucket
    }
}

__global__ void k_compact(const float* __restrict__ scores, unsigned* __restrict__ cnt,
                          float* __restrict__ tvals, int* __restrict__ tidx) {
    int n = blockIdx.x * blockDim.x + threadIdx.x;
    if (n >= N_NODES) return;
    float s = scores[n];
    unsigned b = fkey(s) >> 21;
    unsigned thr = cnt[0];
    if (b > thr) {
        unsigned pos = atomicAdd(&cnt[2], 1u);
        if (pos < KSEL) { tvals[pos] = s; tidx[pos] = n; }
    } else if (b == thr) {
        unsigned q = atomicAdd(&cnt[3], 1u);
        if (q < cnt[4]) {
            unsigned pos = cnt[1] + q;
            if (pos < KSEL) { tvals[pos] = s; tidx[pos] = n; }
        }
    }
}

// bitonic sort of 128 (val,idx) pairs, descending val, ties by idx asc
__global__ void k_sort128(float* __restrict__ tvals, int* __restrict__ tidx) {
    __shared__ float sv[KSEL];
    __shared__ int   si[KSEL];
    int tid = threadIdx.x;
    sv[tid] = tvals[tid]; si[tid] = tidx[tid];
    __syncthreads();
    for (int k = 2; k <= KSEL; k <<= 1) {
        for (int j = k >> 1; j > 0; j >>= 1) {
            int ixj = tid ^ j;
            if (ixj > tid) {
                float v0 = sv[tid], v1 = sv[ixj];
                int i0 = si[tid], i1 = si[ixj];
                bool lt = (v0 < v1) || (v0 == v1 && i0 > i1);
                bool desc = ((tid & k) == 0);
                if (desc ? lt : !lt) {
                    sv[tid] = v1; sv[ixj] = v0;
                    si[tid] = i1; si[ixj] = i0;
                }
            }
            __syncthreads();
        }
    }
    tvals[tid] = sv[tid]; tidx[tid] = si[tid];
}

// xt[d][j] = x[idx[j]][d] * tanh(vals[j])  -> paired layout
__global__ void k_build_xt(const float* __restrict__ x, const float* __restrict__ tvals,
                           const int* __restrict__ tidx, float* __restrict__ xtp) {
    int j = blockIdx.x, d = threadIdx.x;
    int id = tidx[j];
    float tv = tanhf(tvals[j]);
    xtp[pidx(d, j)] = x[(size_t)id * D_IN + d] * tv;
}

// ---------------- GRU: z,r then h,Wn (fp32 WMMA 16x16x4) ----------------
__global__ void k_gru_zr(const float* __restrict__ Wz, const float* __restrict__ Uz,
                         const float* __restrict__ bz,
                         const float* __restrict__ Wr, const float* __restrict__ Ur,
                         const float* __restrict__ br,
                         const float* __restrict__ xtp, const float* __restrict__ Wp,
                         float* __restrict__ zp, float* __restrict__ rp) {
    int wave = threadIdx.x >> 5, lane = threadIdx.x & 31;
    int half = lane >> 4, lrow = lane & 15;
    int i0 = blockIdx.x * 16, j0 = wave * 16;
    int arow = i0 + lrow, bcol = j0 + lrow;
    size_t boff = (size_t)bcol << 1;
    v8f cz = {}; v8f cr = {};
    for (int k0 = 0; k0 < D_IN; k0 += 4) {
        int kk = k0 + 2 * half;
        int kp = (k0 >> 1) + half;
        v2f aZ = *(const v2f*)(Wz + (size_t)arow * D_IN + kk);
        v2f aU = *(const v2f*)(Uz + (size_t)arow * D_IN + kk);
        v2f aR = *(const v2f*)(Wr + (size_t)arow * D_IN + kk);
        v2f aV = *(const v2f*)(Ur + (size_t)arow * D_IN + kk);
        v2f bX = *(const v2f*)(xtp + (size_t)kp * 256 + boff);
        v2f bW = *(const v2f*)(Wp + (size_t)kp * 256 + boff);
        cz = wmma4(aZ, bX, cz);
        cz = wmma4(aU, bW, cz);
        cr = wmma4(aR, bX, cr);
        cr = wmma4(aV, bW, cr);
    }
    for (int v = 0; v < 8; ++v) {
        int row = i0 + v + 8 * half, col = j0 + lrow;
        size_t ob = (size_t)row * D_OUT + col;   // biases are row-major inputs
        size_t op = pidx(row, col);
        zp[op] = 1.f / (1.f + __expf(-(cz[v] + bz[ob])));
        rp[op] = 1.f / (1.f + __expf(-(cr[v] + br[ob])));
    }
}

__global__ void k_gru_h(const float* __restrict__ Wh, const float* __restrict__ Uh,
                        const float* __restrict__ bh,
                        const float* __restrict__ xtp, const float* __restrict__ Wp,
                        const float* __restrict__ zp, const float* __restrict__ rp,
                        float* __restrict__ Wnp) {
    int wave = threadIdx.x >> 5, lane = threadIdx.x & 31;
    int half = lane >> 4, lrow = lane & 15;
    int i0 = blockIdx.x * 16, j0 = wave * 16;
    int arow = i0 + lrow, bcol = j0 + lrow;
    size_t boff = (size_t)bcol << 1;
    v8f ch = {};
    for (int k0 = 0; k0 < D_IN; k0 += 4) {
        int kk = k0 + 2 * half;
        int kp = (k0 >> 1) + half;
        v2f aH = *(const v2f*)(Wh + (size_t)arow * D_IN + kk);
        v2f aU = *(const v2f*)(Uh + (size_t)arow * D_IN + kk);
        v2f bX = *(const v2f*)(xtp + (size_t)kp * 256 + boff);
        v2f rv = *(const v2f*)(rp + (size_t)kp * 256 + boff);
        v2f wv = *(const v2f*)(Wp + (size_t)kp * 256 + boff);
        v2f bRW = rv * wv;
        ch = wmma4(aH, bX, ch);
        ch = wmma4(aU, bRW, ch);
    }
    for (int v = 0; v < 8; ++v) {
        int row = i0 + v + 8 * half, col = j0 + lrow;
        size_t ob = (size_t)row * D_OUT + col;
        size_t op = pidx(row, col);
        float h = tanhf(ch[v] + bh[ob]);
        float z = zp[op], w = Wp[op];
        Wnp[op] = (1.f - z) * w + z * h;
    }
}

// ---------------- y = x @ Wn ----------------
// Wn (paired, LDS row stride 288 -> conflict-free ds_load_b64 across both half-waves)
// A tile (16 x 256, LDS row stride 260 -> conflict-free v2f reads)
#define LDSW_STRIDE 288
#define LDSA_STRIDE 260
#define LDSW_FLOATS (128 * LDSW_STRIDE)
#define LDSA_FLOATS (16 * LDSA_STRIDE)

__global__ void k_gemm_y(const float* __restrict__ x, const float* __restrict__ Wnp,
                         float* __restrict__ y) {
    extern __shared__ float lds[];
    float* ldsW = lds;
    float* ldsA = lds + LDSW_FLOATS;
    int tid = threadIdx.x;
    int m0 = blockIdx.x * 16;

#if defined(__gfx1250__) && __has_builtin(__builtin_amdgcn_global_load_async_to_lds_b128)
    {
        const float* ga = x + (size_t)m0 * D_IN;
        for (int i = tid; i < 128 * 64; i += 256) {
            int kp = i >> 6, q = (i & 63) << 2;
            __builtin_amdgcn_global_load_async_to_lds_b128(
                (ASG v4i*)(Wnp + (size_t)kp * 256 + q),
                (ASL v4i*)(ldsW + kp * LDSW_STRIDE + q), 0, 0);
        }
        for (int i = tid; i < 16 * 64; i += 256) {
            int r = i >> 6, q = (i & 63) << 2;
            __builtin_amdgcn_global_load_async_to_lds_b128(
                (ASG v4i*)(ga + (size_t)r * 256 + q),
                (ASL v4i*)(ldsA + r * LDSA_STRIDE + q), 0, 0);
        }
#if __has_builtin(__builtin_amdgcn_s_wait_asynccnt)
        __builtin_amdgcn_s_wait_asynccnt(0);
#else
        asm volatile("s_wait_asynccnt 0x0" ::: "memory");
#endif
    }
#else
    for (int i = tid; i < 128 * 64; i += 256) {
        int kp = i >> 6, q = (i & 63) << 2;
        *(float4*)(ldsW + kp * LDSW_STRIDE + q) = *(const float4*)(Wnp + kp * 256 + q);
    }
    for (int i = tid; i < 16 * 64; i += 256) {
        int r = i >> 6, q = (i & 63) << 2;
        *(float4*)(ldsA + r * LDSA_STRIDE + q) = *(const float4*)(x + (size_t)(m0 + r) * D_IN + q);
    }
#endif
    __syncthreads();

    int wave = tid >> 5, lane = tid & 31;
    int half = lane >> 4, lrow = lane & 15;
    int j0 = wave * 16;
    int bcol2 = (j0 + lrow) << 1;
    v8f c = {};
    for (int k0 = 0; k0 < D_IN; k0 += 4) {
        int kk = k0 + 2 * half;
        int kp = (k0 >> 1) + half;
        v2f a = *(const v2f*)(ldsA + lrow * LDSA_STRIDE + kk);
        v2f b = *(const v2f*)(ldsW + kp * LDSW_STRIDE + bcol2);
        c = wmma4(a, b, c);
    }
    for (int v = 0; v < 8; ++v)
        y[(size_t)(m0 + v + 8 * half) * D_OUT + j0 + lrow] = c[v];
}

// ---------------- sparse aggregation: agg[dst] += w * y[src] ----------------
__global__ void k_edge_agg(const int* __restrict__ ei, const float* __restrict__ ew,
                           const float* __restrict__ y, float* __restrict__ agg) {
    int wave = threadIdx.x >> 5, lane = threadIdx.x & 31;
    int e = blockIdx.x * 8 + wave;
    int dst = ei[e];
    int src = ei[N_EDGES + e];
    float w = ew[e];
    float4 v = *(const float4*)(y + (size_t)src * D_OUT + lane * 4);
    float* a = agg + (size_t)dst * D_OUT + lane * 4;
    atomicAdd(a + 0, w * v.x);
    atomicAdd(a + 1, w * v.y);
    atomicAdd(a + 2, w * v.z);
    atomicAdd(a + 3, w * v.w);
}

__global__ void k_final(const float4* __restrict__ agg, float4* __restrict__ out) {
    int i = blockIdx.x * 256 + threadIdx.x;
    float4 v = agg[i];
    v.x = v.x >= 0.f ? v.x : SLOPE * v.x;
    v.y = v.y >= 0.f ? v.y : SLOPE * v.y;
    v.z = v.z >= 0.f ? v.z : SLOPE * v.z;
    v.w = v.w >= 0.f ? v.w : SLOPE * v.w;
    out[i] = v;
}

extern "C" void kernel_launch(void* const* d_in, const int* in_sizes, int n_in,
                              void* d_out, int out_size, void* d_ws, size_t ws_size,
                              hipStream_t stream) {
    (void)in_sizes; (void)n_in; (void)out_size; (void)ws_size;
    const float* node_embs   = (const float*)d_in[0];
    const int*   edge_index  = (const int*)d_in[1];
    const float* edge_weight = (const float*)d_in[2];
    const float* mask        = (const float*)d_in[3];
    const float* p   = (const float*)d_in[4];
    const float* Wz  = (const float*)d_in[5];
    const float* Uz  = (const float*)d_in[6];
    const float* bz  = (const float*)d_in[7];
    const float* Wr  = (const float*)d_in[8];
    const float* Ur  = (const float*)d_in[9];
    const float* br  = (const float*)d_in[10];
    const float* Wh  = (const float*)d_in[11];
    const float* Uh  = (const float*)d_in[12];
    const float* bh  = (const float*)d_in[13];
    const float* W0  = (const float*)d_in[14];
    float* out = (float*)d_out;
    float* WS  = (float*)d_ws;

    float*    pn     = WS + OFF_PNORM;
    unsigned* cnt    = (unsigned*)(WS + OFF_CNT);
    unsigned* hist   = (unsigned*)(WS + OFF_HIST);
    float*    scores = WS + OFF_SCORES;
    float*    tvals  = WS + OFF_TVALS;
    int*      tidx   = (int*)(WS + OFF_TIDX);
    float*    xtp    = WS + OFF_XT;
    float*    zp     = WS + OFF_Z;
    float*    rp     = WS + OFF_R;
    float*    Wbuf[2] = { WS + OFF_W0B, WS + OFF_W1B };
    float*    y      = WS + OFF_Y;
    float*    agg    = WS + OFF_AGG;

    k_pnorm<<<1, 256, 0, stream>>>(p, pn);
    k_pack_pairs<<<(D_IN * D_OUT) / 256, 256, 0, stream>>>(W0, Wbuf[0]);

    size_t gemm_lds = (size_t)(LDSW_FLOATS + LDSA_FLOATS) * sizeof(float);

    for (int t = 0; t < T_STEPS; ++t) {
        const float* xT    = node_embs + (size_t)t * N_NODES * D_IN;
        const float* maskT = mask + (size_t)t * N_NODES;
        const int*   eiT   = edge_index + (size_t)t * 2 * N_EDGES;
        const float* ewT   = edge_weight + (size_t)t * N_EDGES;
        float* Wc   = Wbuf[t & 1];
        float* Wn   = Wbuf[(t + 1) & 1];
        float* outT = out + (size_t)t * N_NODES * D_OUT;

        // zero counters + histogram (2064 words)
        k_zero_f4<<<3, 256, 0, stream>>>((float4*)(WS + OFF_CNT), 516);
        k_scores<<<(N_NODES + 7) / 8, 256, 0, stream>>>(xT, p, maskT, pn, scores);
        k_hist<<<(N_NODES + 255) / 256, 256, 0, stream>>>(scores, hist);
        k_select<<<1, 32, 0, stream>>>(hist, cnt);
        k_compact<<<(N_NODES + 255) / 256, 256, 0, stream>>>(scores, cnt, tvals, tidx);
        k_sort128<<<1, KSEL, 0, stream>>>(tvals, tidx);
        k_build_xt<<<KSEL, D_IN, 0, stream>>>(xT, tvals, tidx, xtp);
        k_gru_zr<<<D_IN / 16, 256, 0, stream>>>(Wz, Uz, bz, Wr, Ur, br, xtp, Wc, zp, rp);
        k_gru_h<<<D_IN / 16, 256, 0, stream>>>(Wh, Uh, bh, xtp, Wc, zp, rp, Wn);
        k_zero_f4<<<2500, 256, 0, stream>>>((float4*)agg, (N_NODES * D_OUT) / 4);
        k_gemm_y<<<N_NODES / 16, 256, gemm_lds, stream>>>(xT, Wn, y);
        k_edge_agg<<<N_EDGES / 8, 256, 0, stream>>>(eiT, ewT, y, agg);
        k_final<<<2500, 256, 0, stream>>>((const float4*)agg, (float4*)outT);
    }
}